// Actor_6425271075247
// MI455X (gfx1250) — compile-verified
//
#include <hip/hip_runtime.h>
#include <hip/hip_bf16.h>

// ---------------------------------------------------------------------------
// With DIM=1, LayerNorm over the last axis is exactly its bias (x-mean == 0),
// so _bert(...) == bert.layers[-1].ln2_b broadcast, independent of inputs.
// Whole reference collapses to:
//   colsum[s] = sum_n comp_mask[n,s]            (10 MB stream -> ~0.4us @ HBM)
//   out[s]    = bert2.layers[1].ln2_b * colsum[s]
//   head MLP  : relu(out@W1+b1) -> relu(@W2+b2) -> (wm+bm, tanh-squash(wl+bl))
// Head matmuls run on V_WMMA_F32_16X16X4_F32 (fp32, exact-precision WMMA).
//
// Divergence-free inner loops:
//  * A side: feed every lane A[m][k] = act[k] (rank-1). Only D row 0 is read,
//    and it depends only on A row 0, so duplicated rows are harmless. No
//    (m==0) selects -> no EXEC-mask branches around LDS loads.
//  * B side: out-of-range K rows are multiplied by A==0 (zero-padded act), so
//    only memory safety matters -> clamp the index (min), never guard the
//    load. Unused D columns (stage 3, N>=8) likewise just clamp (m&7).
// ---------------------------------------------------------------------------

#define N_STRUCT 50000
#define SEQ 50
#define HID 256
#define TOTAL (N_STRUCT * SEQ)

typedef __attribute__((ext_vector_type(2))) float v2f;
typedef __attribute__((ext_vector_type(8))) float v8f;

static __device__ __forceinline__ v8f wmma_f32_k4(v2f a, v2f b, v8f c) {
  // 8-arg form: (neg_a, A, neg_b, B, c_mod, C, reuse_a, reuse_b)
  return __builtin_amdgcn_wmma_f32_16x16x4_f32(false, a, false, b, (short)0, c,
                                               false, false);
}

__global__ void zero_ws_kernel(int* p) { p[threadIdx.x] = 0; }

// Column-sum of comp_mask (N_STRUCT x SEQ, int32). Coalesced grid-stride
// stream; 32 consecutive flat indices have distinct (i % 50) -> no
// same-address LDS atomic conflicts within a wave.
__global__ void colsum_kernel(const int* __restrict__ cm,
                              int* __restrict__ partial) {
  __shared__ int bins[64];
  int tid = threadIdx.x;
  if (tid < 64) bins[tid] = 0;
  __syncthreads();
  int stride = gridDim.x * blockDim.x;
  for (int i = blockIdx.x * blockDim.x + tid; i < TOTAL; i += stride) {
    atomicAdd(&bins[i % SEQ], cm[i]);
  }
  __syncthreads();
  if (tid < SEQ) atomicAdd(&partial[tid], bins[tid]);
}

// Single wave32 (EXEC all ones for WMMA): out[s] = c2*colsum[s]; then the
// 50->256->256->(8,8) head MLP on fp32 WMMA.
// Layouts (ISA 7.12.2): A 16x4: lane L -> row m=L%16, x/y = K(2*(L/16))+{0,1}.
// B 4x16: lane L -> col n=L%16, same K split. D: VGPR0 lanes 0-15 = row M=0.
__global__ void head_kernel(const int* __restrict__ colsum,
                            const float* __restrict__ c2p,
                            const float* __restrict__ W1,
                            const float* __restrict__ B1,
                            const float* __restrict__ W2,
                            const float* __restrict__ B2,
                            const float* __restrict__ Wm,
                            const float* __restrict__ Bm,
                            const float* __restrict__ Wl,
                            const float* __restrict__ Bl,
                            float* __restrict__ out) {
  __shared__ float vin[64];
  __shared__ float h1[HID];
  __shared__ float h2[HID];
  const int L = threadIdx.x;          // 0..31
  const int m = L & 15;               // B column within tile
  const int k0 = (L >> 4) << 1;       // lanes 0-15 -> K{0,1}; 16-31 -> K{2,3}
  const float c2 = c2p[0];

  for (int s = L; s < 64; s += 32)
    vin[s] = (s < SEQ) ? c2 * (float)colsum[s] : 0.0f;  // zero-padded K
  __syncthreads();

  // ---- Stage 1: h1 = relu(vin(1x50) @ W1(50x256) + B1) ; K padded to 52
  for (int j = 0; j < 16; ++j) {
    v8f acc = {};
    for (int t = 0; t < 13; ++t) {
      int kA = 4 * t + k0;
      v2f a, b;
      a.x = vin[kA];                  // rank-1 A: same for all lanes
      a.y = vin[kA + 1];
      int r0 = kA < SEQ - 1 ? kA : SEQ - 1;      // clamp address only:
      int r1 = kA + 1 < SEQ ? kA + 1 : SEQ - 1;  // A is 0 for K>=50
      b.x = W1[r0 * HID + j * 16 + m];
      b.y = W1[r1 * HID + j * 16 + m];
      acc = wmma_f32_k4(a, b, acc);
    }
    if (L < 16) {
      float h = acc[0] + B1[j * 16 + L];  // VGPR0 lanes 0-15 = row M=0
      h1[j * 16 + L] = fmaxf(h, 0.0f);
    }
  }
  __syncthreads();

  // ---- Stage 2: h2 = relu(h1(1x256) @ W2(256x256) + B2)
  for (int j = 0; j < 16; ++j) {
    v8f acc = {};
    for (int t = 0; t < 64; ++t) {
      int kA = 4 * t + k0;
      v2f a, b;
      a.x = h1[kA];
      a.y = h1[kA + 1];
      b.x = W2[kA * HID + j * 16 + m];
      b.y = W2[(kA + 1) * HID + j * 16 + m];
      acc = wmma_f32_k4(a, b, acc);
    }
    if (L < 16) {
      float h = acc[0] + B2[j * 16 + L];
      h2[j * 16 + L] = fmaxf(h, 0.0f);
    }
  }
  __syncthreads();

  // ---- Stage 3: mean = h2@Wm + Bm ; log_std = squash(tanh(h2@Wl + Bl))
  const int mc = m & 7;  // clamp column: D columns 8-15 are never read
  v8f am = {};
  v8f al = {};
  for (int t = 0; t < 64; ++t) {
    int kA = 4 * t + k0;
    v2f a, bm, bl;
    a.x = h2[kA];
    a.y = h2[kA + 1];
    bm.x = Wm[kA * 8 + mc];
    bm.y = Wm[(kA + 1) * 8 + mc];
    am = wmma_f32_k4(a, bm, am);
    bl.x = Wl[kA * 8 + mc];
    bl.y = Wl[(kA + 1) * 8 + mc];
    al = wmma_f32_k4(a, bl, al);
  }
  if (L < 8) {
    out[L] = am[0] + Bm[L];
    float tr = tanhf(al[0] + Bl[L]);
    out[8 + L] = -5.0f + 3.5f * (tr + 1.0f);  // LOG_STD_MIN + 0.5*range*(t+1)
  }
}

extern "C" void kernel_launch(void* const* d_in, const int* in_sizes, int n_in,
                              void* d_out, int out_size, void* d_ws,
                              size_t ws_size, hipStream_t stream) {
  // Two plausible pytree flattenings (81 leaves both ways):
  //  insertion order: x, att_mask, comp_mask, bert1(35), bert2(35), head(8)
  //  jax alphabetical: att_mask, bert1, bert2, comp_mask, head, x
  int comp_i, c2_i, w1_i, b1_i, w2_i, b2_i, wm_i, bm_i, wl_i, bl_i;
  bool alpha = (n_in >= 81) && (in_sizes[0] == TOTAL);
  if (alpha) {
    comp_i = 71;
    c2_i = 60;  // bert2.layers[1].ln2_b (sorted: b1,b2,bk,bo,bq,bv,ln1_b,ln1_g,ln2_b)
    b1_i = 72; b2_i = 73; bl_i = 74; bm_i = 75;
    w1_i = 76; w2_i = 77; wl_i = 78; wm_i = 79;
  } else {
    comp_i = 2;
    c2_i = 72;  // 3 + 35 + pos,ln_g,ln_b + layer0(16) + 15 -> layers[1].ln2_b
    w1_i = 73; b1_i = 74; w2_i = 75; b2_i = 76;
    wm_i = 77; bm_i = 78; wl_i = 79; bl_i = 80;
  }

  int* partial = (int*)d_ws;  // 64 ints of scratch

  zero_ws_kernel<<<1, 64, 0, stream>>>(partial);
  colsum_kernel<<<1024, 256, 0, stream>>>((const int*)d_in[comp_i], partial);
  head_kernel<<<1, 32, 0, stream>>>(
      partial, (const float*)d_in[c2_i], (const float*)d_in[w1_i],
      (const float*)d_in[b1_i], (const float*)d_in[w2_i],
      (const float*)d_in[b2_i], (const float*)d_in[wm_i],
      (const float*)d_in[bm_i], (const float*)d_in[wl_i],
      (const float*)d_in[bl_i], (float*)d_out);
}